// SlowNetB_33371895890545
// MI455X (gfx1250) — compile-verified
//
#include <hip/hip_runtime.h>

// Fast-weight Hebbian scan for MI455X (gfx1250, wave32).
//
// Phase 1 (WMMA): Z0 = XS @ WA, Z1 = XS @ WB  -- two 512^3 f32 GEMMs via
//                 v_wmma_f32_16x16x4_f32 (full precision; phase is <1% of work).
// Phase 2 (scan): fw[i][j] recurrence is element-local:
//                 fw_t = tanh(fw_{t-1} + eta*Z0[t][i]*Z1[t][j])
//                 y[t][j] = sum_i xs[t][i]*fw_t[i][j]
//                 One wave per column j; lane owns 16 rows -> the whole 1MB fw
//                 state lives in VGPRs (512 waves * 32 lanes * 16 regs * 4B).
//                 tanh uses the gfx1250 hardware v_tanh_f32 (TRANS op).

#define DD 512
#define TT 512

typedef __attribute__((ext_vector_type(2))) float v2f;
typedef __attribute__((ext_vector_type(8))) float v8f;

// ---------------------------------------------------------------------------
// Phase 1: Z0 = XS @ WA, Z1 = XS @ WB with V_WMMA_F32_16X16X4_F32.
// Tile = 16x16 of C per wave; K advances 4 per WMMA (128 iterations).
// A 16x4 layout (ISA 7.12.2): lane L holds M = L&15, VGPR0/1 = K = 2*(L>>4)+{0,1}.
// B 4x16 layout (row striped across lanes): lane L holds N = L&15,
//   VGPR0/1 = K = 2*(L>>4)+{0,1}.
// C/D 16x16 f32: VGPR r -> M = r + 8*(L>>4), N = L&15.
// ---------------------------------------------------------------------------
__global__ __launch_bounds__(256) void fw_gemm_wmma(
    const float* __restrict__ xs, const float* __restrict__ wa,
    const float* __restrict__ wb, float* __restrict__ z0out,
    float* __restrict__ z1out)
{
    const int lane = threadIdx.x & 31;
    const int wid  = threadIdx.x >> 5;

    int tile = blockIdx.x * 8 + wid;      // 0..2047 : [0,1024) -> Z0, [1024,2048) -> Z1
    const int which = tile >> 10;
    tile &= 1023;
    const int tm = tile >> 5;             // 0..31
    const int tn = tile & 31;             // 0..31

    const float* __restrict__ B = which ? wb : wa;
    float*       __restrict__ C = which ? z1out : z0out;

    const int lo = lane & 15;
    const int hi = lane >> 4;             // 0 or 1
    const int m  = tm * 16 + lo;
    const int n  = tn * 16 + lo;

    v8f acc = {};
    for (int k0 = 0; k0 < DD; k0 += 4) {
        const int ka = k0 + 2 * hi;
        // A fragment: two consecutive K values of row m (8B aligned).
        v2f a = *(const v2f*)(xs + m * DD + ka);
        // B fragment: two K rows of column n.
        v2f b;
        b.x = B[(ka + 0) * DD + n];
        b.y = B[(ka + 1) * DD + n];
        acc = __builtin_amdgcn_wmma_f32_16x16x4_f32(
            /*neg_a=*/false, a, /*neg_b=*/false, b,
            /*c_mod=*/(short)0, acc, /*reuse_a=*/false, /*reuse_b=*/false);
    }

#pragma unroll
    for (int r = 0; r < 8; ++r) {
        C[(tm * 16 + r + 8 * hi) * DD + n] = acc[r];
    }
}

// ---------------------------------------------------------------------------
// Phase 2: per-element tanh recurrence + in-wave dot-product reduction.
// ---------------------------------------------------------------------------
__device__ __forceinline__ float tanh_hw(float x) {
    float r;
    asm volatile("v_tanh_f32 %0, %1" : "=v"(r) : "v"(x));
    return r;
}

__global__ __launch_bounds__(256) void fw_scan(
    const float* __restrict__ xs, const float* __restrict__ fw_init,
    const float* __restrict__ eta_p, const float* __restrict__ z0,
    const float* __restrict__ z1, float* __restrict__ y)
{
    const int lane = threadIdx.x & 31;
    const int wid  = threadIdx.x >> 5;
    const int j    = blockIdx.x * 8 + wid;  // output column, 0..511
    const int i0   = lane * 16;             // this lane's first row

    const float eta = eta_p[0];

    // Fast-weight column slice lives in registers for the whole scan.
    float fw[16];
#pragma unroll
    for (int k = 0; k < 16; ++k)
        fw[k] = fw_init[(i0 + k) * DD + j];

    for (int t = 0; t < TT; ++t) {
        const float c = eta * z1[t * DD + j];   // broadcast load (same addr per wave)

        const float4* __restrict__ z0p = (const float4*)(z0 + t * DD + i0);
        const float4* __restrict__ xsp = (const float4*)(xs + t * DD + i0);

        if (t + 1 < TT) {                       // global_prefetch_b8 of next rows
            __builtin_prefetch(z0 + (t + 1) * DD + i0, 0, 3);
            __builtin_prefetch(xs + (t + 1) * DD + i0, 0, 3);
        }

        float zr[16], xr[16];
#pragma unroll
        for (int q = 0; q < 4; ++q) {           // 2x 64B coalesced per lane
            float4 zv = z0p[q];
            float4 xv = xsp[q];
            zr[q * 4 + 0] = zv.x; zr[q * 4 + 1] = zv.y;
            zr[q * 4 + 2] = zv.z; zr[q * 4 + 3] = zv.w;
            xr[q * 4 + 0] = xv.x; xr[q * 4 + 1] = xv.y;
            xr[q * 4 + 2] = xv.z; xr[q * 4 + 3] = xv.w;
        }

        // 16 independent tanh recurrences (TRANS ops back-to-back co-execute).
#pragma unroll
        for (int k = 0; k < 16; ++k)
            fw[k] = tanh_hw(fmaf(c, zr[k], fw[k]));

        // Partial dot product over this lane's 16 rows.
        float partial = 0.0f;
#pragma unroll
        for (int k = 0; k < 16; ++k)
            partial = fmaf(xr[k], fw[k], partial);

        // Wave32 xor-tree reduction over the 32 lanes (512 rows total).
#pragma unroll
        for (int off = 16; off >= 1; off >>= 1)
            partial += __shfl_xor(partial, off, 32);

        if (lane == 0)
            y[t * DD + j] = partial;
    }
}

// ---------------------------------------------------------------------------
// Launch: phase-1 GEMMs then the scan; stream ordering is the only sync needed.
// ---------------------------------------------------------------------------
extern "C" void kernel_launch(void* const* d_in, const int* in_sizes, int n_in,
                              void* d_out, int out_size, void* d_ws, size_t ws_size,
                              hipStream_t stream) {
    (void)in_sizes; (void)n_in; (void)out_size; (void)ws_size;

    const float* xs  = (const float*)d_in[0];  // [T, D]
    const float* wa  = (const float*)d_in[1];  // [D, D]
    const float* wb  = (const float*)d_in[2];  // [D, D]
    const float* fw0 = (const float*)d_in[3];  // [D, D]
    const float* eta = (const float*)d_in[4];  // scalar

    float* y  = (float*)d_out;                 // [T, D]
    float* z0 = (float*)d_ws;                  // [T, D] scratch (1 MB)
    float* z1 = z0 + (size_t)TT * DD;          // [T, D] scratch (1 MB)

    // 2048 tiles / 8 waves per block = 256 blocks.
    fw_gemm_wmma<<<256, 256, 0, stream>>>(xs, wa, wb, z0, z1);
    // 512 columns / 8 waves per block = 64 blocks.
    fw_scan<<<64, 256, 0, stream>>>(xs, fw0, eta, z0, z1, y);
}